// MAA_89000312308386
// MI455X (gfx1250) — compile-verified
//
#include <hip/hip_runtime.h>

typedef __attribute__((ext_vector_type(2))) float v2f;
typedef __attribute__((ext_vector_type(8))) float v8f;

#define NN 4096
#define BPAIRS 4096
#define EPSF 1e-16f

// ---------------------------------------------------------------------------
// Kernel 1: bandwidth-bound gather + 6-way fused reduction.
// One 256-thread block per (lid, pair). Each thread owns 16 consecutive j's
// (4 x float4 / b128 loads per stream, 6 streams). A1+A2 (128MB) are
// L2-resident (192MB L2), so repeat row gathers hit L2, not HBM.
// ---------------------------------------------------------------------------
__global__ __launch_bounds__(256) void pair_reduce_kernel(
    const float* __restrict__ A1, const float* __restrict__ A2,
    const float* __restrict__ f1, const float* __restrict__ f2,
    const int* __restrict__ is1, const int* __restrict__ id1,
    const int* __restrict__ is2, const int* __restrict__ id2,
    float* __restrict__ xout)
{
    const int p   = blockIdx.x;          // 0..8191
    const int lid = p >> 12;
    const int b   = p & (BPAIRS - 1);

    const float* A  = lid ? A2 : A1;
    const float* Ao = lid ? A1 : A2;
    const float* fa = lid ? f2 : f1;
    const float* fb = lid ? f1 : f2;
    const int*   si = lid ? is2 : is1;
    const int*   di = lid ? id2 : id1;

    const int isrc = si[b];
    const int idst = di[b];

    const float4* rAs  = (const float4*)(A  + (size_t)isrc * NN);
    const float4* rAod = (const float4*)(Ao + (size_t)idst * NN);
    const float4* rAos = (const float4*)(Ao + (size_t)isrc * NN);
    const float4* rAd  = (const float4*)(A  + (size_t)idst * NN);
    const float4* fav  = (const float4*)fa;
    const float4* fbv  = (const float4*)fb;

    const int t = threadIdx.x;
    float s1 = 0.f, s2 = 0.f, n1 = 0.f, n2 = 0.f, n3 = 0.f, n4 = 0.f;

#define ACC(c)                                                              \
    {                                                                       \
        float t1 = as.c * fa4.c;  float t2 = aod.c * fb4.c;                 \
        float t3 = aos.c * fb4.c; float t4 = ad.c  * fa4.c;                 \
        s1 = fmaf(t1, t2, s1);    s2 = fmaf(t3, t4, s2);                    \
        float u1 = t1 + EPSF, u2 = t2 + EPSF, u3 = t3 + EPSF, u4 = t4 + EPSF; \
        n1 = fmaf(u1, u1, n1);    n2 = fmaf(u2, u2, n2);                    \
        n3 = fmaf(u3, u3, n3);    n4 = fmaf(u4, u4, n4);                    \
    }

#pragma unroll
    for (int k = 0; k < 4; ++k) {
        const int q = t * 4 + k;                 // float4 index
        float4 as  = rAs[q];
        float4 aod = rAod[q];
        float4 aos = rAos[q];
        float4 ad  = rAd[q];
        float4 fa4 = fav[q];
        float4 fb4 = fbv[q];
        ACC(x) ACC(y) ACC(z) ACC(w)
    }
#undef ACC

    // wave32 shuffle reduction of the 6 accumulators
#pragma unroll
    for (int m = 16; m >= 1; m >>= 1) {
        s1 += __shfl_xor(s1, m, 32);
        s2 += __shfl_xor(s2, m, 32);
        n1 += __shfl_xor(n1, m, 32);
        n2 += __shfl_xor(n2, m, 32);
        n3 += __shfl_xor(n3, m, 32);
        n4 += __shfl_xor(n4, m, 32);
    }

    __shared__ float red[8][6];
    const int wave = t >> 5, lane = t & 31;
    if (lane == 0) {
        red[wave][0] = s1; red[wave][1] = s2;
        red[wave][2] = n1; red[wave][3] = n2;
        red[wave][4] = n3; red[wave][5] = n4;
    }
    __syncthreads();
    if (t == 0) {
        float S1 = 0.f, S2 = 0.f, N1 = 0.f, N2 = 0.f, N3 = 0.f, N4 = 0.f;
#pragma unroll
        for (int w = 0; w < 8; ++w) {
            S1 += red[w][0]; S2 += red[w][1];
            N1 += red[w][2]; N2 += red[w][3];
            N3 += red[w][4]; N4 += red[w][5];
        }
        const float D    = sqrtf(N1) * sqrtf(N2);
        const float Drev = sqrtf(N3) * sqrtf(N4);
        xout[p] = S1 / D + S2 / Drev;
    }
}

// ---------------------------------------------------------------------------
// Kernel 2: the two 1->16->1 MLPs via V_WMMA_F32_16X16X4_F32.
// One wave handles 16 pairs: A[m,0]=x[m], B[0,n]=w1[n], C[m,n]=b1[n]
//   -> D[m,n] = x[m]*w1[n] + b1[n]   (exact f32, matches reference math)
// ReLU, scale by w2[n], contract over n via per-wave LDS, add b2; repeat for
// the second MLP; store.
// ---------------------------------------------------------------------------
__global__ __launch_bounds__(256) void mlp_wmma_kernel(
    const float* __restrict__ xin,
    const float* __restrict__ g1w1, const float* __restrict__ g1b1,
    const float* __restrict__ g1w2, const float* __restrict__ g1b2,
    const float* __restrict__ g2w1, const float* __restrict__ g2b1,
    const float* __restrict__ g2w2, const float* __restrict__ g2b2,
    float* __restrict__ out)
{
    __shared__ float lds[8 * 256];

    const int t    = threadIdx.x;
    const int wave = t >> 5;
    const int lane = t & 31;
    const int tile = blockIdx.x * 8 + wave;   // 512 tiles of 16 pairs
    const int p0   = tile * 16;
    const int lid  = p0 >> 12;                // 4096 % 16 == 0, uniform per tile
    const int n    = lane & 15;
    const bool lo  = (lane < 16);

    float* L = &lds[wave * 256];
    const int mbase = lo ? 0 : 8;             // C/D rows held by this lane half

    // ---- layer g1 ----
    const float xv  = xin[p0 + n];
    const float w1v = g1w1[lid * 16 + n];
    const float b1v = g1b1[lid * 16 + n];
    const float w2v = g1w2[lid * 16 + n];

    v2f Am = { lo ? xv  : 0.0f, 0.0f };       // A[m,0]=x[m], other K zero
    v2f Bm = { lo ? w1v : 0.0f, 0.0f };       // B[0,n]=w1[n], other K zero
    v8f C;
#pragma unroll
    for (int r = 0; r < 8; ++r) C[r] = b1v;   // C[m,n] = b1[n]

    v8f D = __builtin_amdgcn_wmma_f32_16x16x4_f32(
        false, Am, false, Bm, (short)0, C, false, false);

#pragma unroll
    for (int r = 0; r < 8; ++r)               // relu, pre-scale by w2[n]
        L[(mbase + r) * 16 + n] = fmaxf(D[r], 0.0f) * w2v;
    __syncthreads();

    float y1 = 0.0f;
#pragma unroll
    for (int j = 0; j < 16; ++j) y1 += L[n * 16 + j];   // row n sum
    y1 += g1b2[lid];                                    // BETA=1, /1.0
    __syncthreads();

    // ---- layer g2 ----
    const float v1v = g2w1[lid * 16 + n];
    const float c1v = g2b1[lid * 16 + n];
    const float v2v = g2w2[lid * 16 + n];

    v2f Am2 = { lo ? y1  : 0.0f, 0.0f };
    v2f Bm2 = { lo ? v1v : 0.0f, 0.0f };
    v8f C2;
#pragma unroll
    for (int r = 0; r < 8; ++r) C2[r] = c1v;

    v8f D2 = __builtin_amdgcn_wmma_f32_16x16x4_f32(
        false, Am2, false, Bm2, (short)0, C2, false, false);

#pragma unroll
    for (int r = 0; r < 8; ++r)
        L[(mbase + r) * 16 + n] = fmaxf(D2[r], 0.0f) * v2v;
    __syncthreads();

    float y2 = 0.0f;
#pragma unroll
    for (int j = 0; j < 16; ++j) y2 += L[n * 16 + j];
    y2 += g2b2[lid];

    if (lo) out[p0 + n] = y2;                 // out shape (2,4096,1) flat
}

// ---------------------------------------------------------------------------
extern "C" void kernel_launch(void* const* d_in, const int* in_sizes, int n_in,
                              void* d_out, int out_size, void* d_ws, size_t ws_size,
                              hipStream_t stream)
{
    const float* A1  = (const float*)d_in[0];
    const float* A2  = (const float*)d_in[1];
    const float* f1  = (const float*)d_in[2];
    const float* f2  = (const float*)d_in[3];
    const int*   is1 = (const int*)d_in[4];
    const int*   id1 = (const int*)d_in[5];
    const int*   is2 = (const int*)d_in[6];
    const int*   id2 = (const int*)d_in[7];
    const float* g1w1 = (const float*)d_in[8];
    const float* g1b1 = (const float*)d_in[9];
    const float* g1w2 = (const float*)d_in[10];
    const float* g1b2 = (const float*)d_in[11];
    const float* g2w1 = (const float*)d_in[12];
    const float* g2b1 = (const float*)d_in[13];
    const float* g2w2 = (const float*)d_in[14];
    const float* g2b2 = (const float*)d_in[15];

    float* xbuf = (float*)d_ws;               // 8192 floats of scratch

    pair_reduce_kernel<<<2 * BPAIRS, 256, 0, stream>>>(
        A1, A2, f1, f2, is1, id1, is2, id2, xbuf);

    mlp_wmma_kernel<<<64, 256, 0, stream>>>(
        xbuf, g1w1, g1b1, g1w2, g1b2, g2w1, g2b1, g2w2, g2b2,
        (float*)d_out);
}